// CellMemoryGraph_83330955477678
// MI455X (gfx1250) — compile-verified
//
#include <hip/hip_runtime.h>
#include <math.h>

// Problem constants (from reference)
#define NCC    64
#define CCn    256
#define DDim   32
#define AALPHA 16
#define BBord  16
#define HMODD  32
#define NNtot  (NCC*CCn)   // 16384
#define BSS    8
#define MOD_IN  113
#define MOD_OUT 65

typedef __attribute__((ext_vector_type(2))) float v2f;
typedef __attribute__((ext_vector_type(8))) float v8f;

static __device__ __forceinline__ v8f wmma4(v2f a, v2f b, v8f c) {
  // V_WMMA_F32_16X16X4_F32 : D(16x16 f32) = A(16x4 f32) * B(4x16 f32) + C
  return __builtin_amdgcn_wmma_f32_16x16x4_f32(false, a, false, b, (short)0, c, false, false);
}
static __device__ __forceinline__ float sigm(float x) { return 1.0f / (1.0f + __expf(-x)); }
static __device__ __forceinline__ float ntload(const float* p) {
  return __builtin_nontemporal_load(p);
}

// ---------------- K0: h injection -> h_upd ----------------
__global__ __launch_bounds__(256) void k0_inject(const float* __restrict__ h,
                                                 const float* __restrict__ x,
                                                 float* __restrict__ h_upd) {
  size_t i = (size_t)blockIdx.x * blockDim.x + threadIdx.x;  // BS*NC*C*D = 4194304
  int d  = (int)(i & 31);
  int c  = (int)((i >> 5) & 255);
  int nc = (int)((i >> 13) & 63);
  int b  = (int)(i >> 19);
  float v = h[i];
  if (c < AALPHA) v += x[(b * NCC + nc) * DDim + d];
  h_upd[i] = v;
}

// ---------------- K1: per-neuron mod MLP (f32 WMMA, NT weight stream) ----------------
__global__ __launch_bounds__(32) void k1_mod(
    const float* __restrict__ h_upd, const float* __restrict__ heb,
    const float* __restrict__ decay, const float* __restrict__ prim,
    const float* __restrict__ nid,
    const float* __restrict__ w1, const float* __restrict__ b1,
    const float* __restrict__ w2, const float* __restrict__ b2,
    float* __restrict__ mod_out) {
  __shared__ float s_in[16 * 116];   // batch rows (pad 8->16), input cols (pad 113->116, zeros)
  __shared__ float s_hid[16 * 32];
  const int n  = blockIdx.x;         // neuron id
  const int nc = n >> 8, c = n & 255;
  const int lane = threadIdx.x;
  const size_t cellrow = (size_t)nc * CCn + c;

  // ---- branch-free input staging: zero-fill, then per-source coalesced ranges ----
  for (int t = lane; t < 16 * 116; t += 32) s_in[t] = 0.0f;
  for (int t = lane; t < BSS * 16; t += 32) {         // heb -> cols 0..15
    int row = t >> 4, col = t & 15;
    s_in[row * 116 + col] = heb[((size_t)row * NNtot + cellrow) * 16 + col];
  }
  for (int t = lane; t < BSS * 32; t += 32) {         // h -> cols 16..47
    int row = t >> 5, d = t & 31;
    s_in[row * 116 + 16 + d] = h_upd[((size_t)row * NNtot + cellrow) * 32 + d];
  }
  if (lane < BSS)                                      // decay -> col 48
    s_in[lane * 116 + 48] = decay[(size_t)lane * NNtot + cellrow];
  for (int t = lane; t < BSS * 32; t += 32) {         // prim -> cols 49..80
    int row = t >> 5, d = t & 31;
    s_in[row * 116 + 49 + d] = prim[((size_t)row * NNtot + cellrow) * 32 + d];
  }
  {                                                    // nid -> cols 81..112 (batch-invariant)
    float v = nid[cellrow * 32 + lane];
    #pragma unroll
    for (int row = 0; row < BSS; ++row) s_in[row * 116 + 81 + lane] = v;
  }
  __syncthreads();

  const int half = lane >> 4, lp = lane & 15;
  // Hoisted per-lane base pointers: all loop loads use immediate offsets.
  const float* W1   = w1 + (size_t)n * (HMODD * MOD_IN);
  const float* row0 = W1 + lp * MOD_IN + 2 * half;         // B rows for N-tile 0
  const float* row1 = W1 + (16 + lp) * MOD_IN + 2 * half;  // B rows for N-tile 1
  const float* arow = s_in + lp * 116 + 2 * half;          // A row in LDS

  v8f z = {};
  v8f acc0 = z, acc1 = z;
  // Main K loop: k in [0,112) — no clamping needed, constant offsets after unroll.
  #pragma unroll
  for (int k0 = 0; k0 < 112; k0 += 4) {
    v2f a;   a[0]   = arow[k0];          a[1]   = arow[k0 + 1];
    v2f bb0; bb0[0] = ntload(row0 + k0); bb0[1] = ntload(row0 + k0 + 1);
    v2f bb1; bb1[0] = ntload(row1 + k0); bb1[1] = ntload(row1 + k0 + 1);
    acc0 = wmma4(a, bb0, acc0);
    acc1 = wmma4(a, bb1, acc1);
  }
  {  // Tail chunk k0=112: A is zero for k>=113, clamp B addresses in-bounds.
    int ka = min(112 + 2 * half, MOD_IN - 1) - 2 * half;       // relative to row base
    int kb = min(113 + 2 * half, MOD_IN - 1) - 2 * half;
    v2f a;   a[0]   = arow[112];          a[1]   = arow[113];
    v2f bb0; bb0[0] = ntload(row0 + ka);  bb0[1] = ntload(row0 + kb);
    v2f bb1; bb1[0] = ntload(row1 + ka);  bb1[1] = ntload(row1 + kb);
    acc0 = wmma4(a, bb0, acc0);
    acc1 = wmma4(a, bb1, acc1);
  }
  float bias0 = b1[n * 32 + lp], bias1 = b1[n * 32 + 16 + lp];
  #pragma unroll
  for (int j = 0; j < 8; ++j) {
    int m = j + 8 * half;
    s_hid[m * 32 + lp]      = tanhf(acc0[j] + bias0);
    s_hid[m * 32 + 16 + lp] = tanhf(acc1[j] + bias1);
  }
  __syncthreads();

  const float* W2 = w2 + (size_t)n * (HMODD * MOD_OUT);
  // Column clamp is k-invariant: hoist 5 clamped column base pointers.
  const float* w2c[5];
  #pragma unroll
  for (int t = 0; t < 5; ++t)
    w2c[t] = W2 + (2 * half) * MOD_OUT + min(t * 16 + lp, MOD_OUT - 1);
  const float* hrow = s_hid + lp * 32 + 2 * half;

  v8f outv[5];
  #pragma unroll
  for (int t = 0; t < 5; ++t) outv[t] = z;
  #pragma unroll
  for (int k0 = 0; k0 < 32; k0 += 4) {
    v2f a;  a[0] = hrow[k0];  a[1] = hrow[k0 + 1];
    #pragma unroll
    for (int t = 0; t < 5; ++t) {
      v2f bb;
      bb[0] = ntload(w2c[t] + k0 * MOD_OUT);
      bb[1] = ntload(w2c[t] + (k0 + 1) * MOD_OUT);
      outv[t] = wmma4(a, bb, outv[t]);
    }
  }
  if (half == 0) {   // rows m=0..7 are the real batches
    #pragma unroll
    for (int t = 0; t < 5; ++t) {
      int col = t * 16 + lp;
      if (col < MOD_OUT) {
        float bo = b2[n * MOD_OUT + col];
        #pragma unroll
        for (int j = 0; j < 8; ++j)
          mod_out[((size_t)j * NNtot + n) * MOD_OUT + col] = outv[t][j] + bo;
      }
    }
  }
}

// ---------------- K2: neighbor aggregation (one wave per (b,n,c)) ----------------
__global__ __launch_bounds__(256) void k2_agg(
    const float* __restrict__ mod_out, const float* __restrict__ heb,
    const float* __restrict__ hebb, const float* __restrict__ prev,
    const int* __restrict__ conn, const int* __restrict__ bconn,
    float* __restrict__ agg) {
  int gid = blockIdx.x * blockDim.x + threadIdx.x;
  int wid = gid >> 5, lane = gid & 31;
  int b  = wid >> 14;
  int nl = wid & 16383;              // nc*256 + c
  int nc = nl >> 8, c = nl & 255;
  const float* mo = mod_out + (size_t)wid * MOD_OUT;

  float wk = 0.0f; int ik = 0;
  if (lane < 16) {
    wk = sigm(mo[lane] + heb[(size_t)wid * 16 + lane]);
    ik = conn[nl * 16 + lane];
  }
  float acc = 0.0f;
  const size_t cellbase = (size_t)b * NNtot + nc * CCn;
  #pragma unroll
  for (int k = 0; k < 16; ++k) {
    float w = __shfl(wk, k);
    int idx = __shfl(ik, k);
    acc += w * prev[(cellbase + idx) * DDim + lane];
  }
  if (c >= AALPHA && c < AALPHA + BBord) {
    int cb = c - AALPHA;
    float wbk = 0.0f; int jk = 0;
    if (lane < 16) {
      wbk = sigm(mo[16 + lane] + hebb[((size_t)(b * NCC + nc) * BBord + cb) * 16 + lane]);
      jk  = bconn[(nc * BBord + cb) * 16 + lane];
    }
    #pragma unroll
    for (int k = 0; k < 16; ++k) {
      float w = __shfl(wbk, k);
      int j  = __shfl(jk, k);
      acc += w * prev[((size_t)b * NNtot + (j >> 4) * CCn + AALPHA + (j & 15)) * DDim + lane];
    }
  }
  agg[(size_t)wid * DDim + lane] = acc;
}

// ---------------- K3: shared state/msg MLPs (LDS-staged weights, f32 WMMA) --------
__global__ __launch_bounds__(256) void k3_mlps(
    const float* __restrict__ h_upd, const float* __restrict__ aggb,
    const float* __restrict__ mod_out, const float* __restrict__ nid,
    const float* __restrict__ sw1, const float* __restrict__ sb1,
    const float* __restrict__ sw2, const float* __restrict__ sb2,
    const float* __restrict__ mw1, const float* __restrict__ mb1,
    const float* __restrict__ mw2, const float* __restrict__ mb2,
    float* __restrict__ out_hnew, float* __restrict__ out_msg) {
  extern __shared__ float sm[];
  const int W1O = 0, W2O = 12416, MW1O = 16512, MW2O = 28800,
            B1O = 32896, B2O = 33024, MB1O = 33056, MB2O = 33184, WAVEO = 33216;
  int tid = threadIdx.x;
  for (int t = tid; t < 12416; t += 256) sm[W1O + t]  = sw1[t];  // (128,97)
  for (int t = tid; t < 4096;  t += 256) sm[W2O + t]  = sw2[t];  // (32,128)
  for (int t = tid; t < 12288; t += 256) sm[MW1O + t] = mw1[t];  // (128,96)
  for (int t = tid; t < 4096;  t += 256) sm[MW2O + t] = mw2[t];  // (32,128)
  if (tid < 128) sm[B1O + tid]  = sb1[tid];
  if (tid < 32)  sm[B2O + tid]  = sb2[tid];
  if (tid < 128) sm[MB1O + tid] = mb1[tid];
  if (tid < 32)  sm[MB2O + tid] = mb2[tid];
  __syncthreads();

  int wave = tid >> 5, lane = tid & 31;
  int half = lane >> 4, lp = lane & 15;
  float* io  = sm + WAVEO + wave * 3712;   // 16 x 100 input tile
  float* hid = io + 1600;                  // 16 x 128 hidden tile
  int tt  = blockIdx.x * 8 + wave;         // row-tile id, 8192 total (exact)
  int r0  = tt * 16;
  int nl0 = r0 & 16383;

  // ---- branch-free state-input staging: [h(32) | agg(32) | new_prim(32) | new_decay(1) | 0 pad] ----
  for (int t = lane; t < 512; t += 32) {
    int m = t >> 5, d = t & 31;
    io[m * 100 + d] = h_upd[(size_t)(r0 + m) * 32 + d];
  }
  for (int t = lane; t < 512; t += 32) {
    int m = t >> 5, d = t & 31;
    io[m * 100 + 32 + d] = aggb[(size_t)(r0 + m) * 32 + d];
  }
  for (int t = lane; t < 512; t += 32) {
    int m = t >> 5, d = t & 31;
    io[m * 100 + 64 + d] = mod_out[(size_t)(r0 + m) * 65 + 33 + d];
  }
  if (lane < 16) {
    io[lane * 100 + 96] = mod_out[(size_t)(r0 + lane) * 65 + 32];
    io[lane * 100 + 97] = 0.0f;
    io[lane * 100 + 98] = 0.0f;
    io[lane * 100 + 99] = 0.0f;
  }
  __syncthreads();

  v8f z = {};
  // ---- state layer 1: (16x97) x (97x128) ----
  v8f a1[8];
  #pragma unroll
  for (int t = 0; t < 8; ++t) a1[t] = z;
  for (int k0 = 0; k0 < 100; k0 += 4) {
    int k = k0 + 2 * half;
    v2f a;  a[0] = io[lp * 100 + k];  a[1] = io[lp * 100 + k + 1];
    #pragma unroll
    for (int t = 0; t < 8; ++t) {
      int hcol = t * 16 + lp;
      v2f bb;  bb[0] = sm[W1O + hcol * 97 + k];  bb[1] = sm[W1O + hcol * 97 + k + 1];
      a1[t] = wmma4(a, bb, a1[t]);
    }
  }
  #pragma unroll
  for (int t = 0; t < 8; ++t) {
    float bv = sm[B1O + t * 16 + lp];
    #pragma unroll
    for (int j = 0; j < 8; ++j)
      hid[(j + 8 * half) * 128 + t * 16 + lp] = tanhf(a1[t][j] + bv);
  }
  __syncthreads();

  // ---- state layer 2: (16x128) x (128x32) ----
  v8f d0 = z, d1 = z;
  for (int k0 = 0; k0 < 128; k0 += 4) {
    int k = k0 + 2 * half;
    v2f a;   a[0] = hid[lp * 128 + k];   a[1] = hid[lp * 128 + k + 1];
    v2f b0;  b0[0] = sm[W2O + lp * 128 + k];         b0[1] = sm[W2O + lp * 128 + k + 1];
    v2f b1v; b1v[0] = sm[W2O + (16 + lp) * 128 + k]; b1v[1] = sm[W2O + (16 + lp) * 128 + k + 1];
    d0 = wmma4(a, b0, d0);
    d1 = wmma4(a, b1v, d1);
  }
  // h_new directly from accumulators; stage into msg-input cols 0..31
  float sbias0 = sm[B2O + lp], sbias1 = sm[B2O + 16 + lp];
  #pragma unroll
  for (int j = 0; j < 8; ++j) {
    int m = j + 8 * half;
    size_t r = (size_t)r0 + m;
    float dec = sigm(mod_out[r * 65 + 32]);
    float h0 = h_upd[r * 32 + lp];
    float h1 = h_upd[r * 32 + 16 + lp];
    float hn0 = dec * h0 + (1.0f - dec) * tanhf(d0[j] + sbias0);
    float hn1 = dec * h1 + (1.0f - dec) * tanhf(d1[j] + sbias1);
    out_hnew[r * 32 + lp]      = hn0;
    out_hnew[r * 32 + 16 + lp] = hn1;
    io[m * 100 + lp]      = hn0;
    io[m * 100 + 16 + lp] = hn1;
  }
  // msg input: cols 32..63 (agg) already in place; overwrite cols 64..95 with nid, zero col 96
  for (int t = lane; t < 512; t += 32) {
    int m = t >> 5, d = t & 31;
    io[m * 100 + 64 + d] = nid[(size_t)(nl0 + m) * 32 + d];
  }
  if (lane < 16) io[lane * 100 + 96] = 0.0f;
  __syncthreads();

  // ---- msg layer 1: (16x96) x (96x128) ----
  #pragma unroll
  for (int t = 0; t < 8; ++t) a1[t] = z;
  for (int k0 = 0; k0 < 96; k0 += 4) {
    int k = k0 + 2 * half;
    v2f a;  a[0] = io[lp * 100 + k];  a[1] = io[lp * 100 + k + 1];
    #pragma unroll
    for (int t = 0; t < 8; ++t) {
      int hcol = t * 16 + lp;
      v2f bb;  bb[0] = sm[MW1O + hcol * 96 + k];  bb[1] = sm[MW1O + hcol * 96 + k + 1];
      a1[t] = wmma4(a, bb, a1[t]);
    }
  }
  #pragma unroll
  for (int t = 0; t < 8; ++t) {
    float bv = sm[MB1O + t * 16 + lp];
    #pragma unroll
    for (int j = 0; j < 8; ++j)
      hid[(j + 8 * half) * 128 + t * 16 + lp] = tanhf(a1[t][j] + bv);
  }
  __syncthreads();

  // ---- msg layer 2 ----
  d0 = z; d1 = z;
  for (int k0 = 0; k0 < 128; k0 += 4) {
    int k = k0 + 2 * half;
    v2f a;   a[0] = hid[lp * 128 + k];   a[1] = hid[lp * 128 + k + 1];
    v2f b0;  b0[0] = sm[MW2O + lp * 128 + k];         b0[1] = sm[MW2O + lp * 128 + k + 1];
    v2f b1v; b1v[0] = sm[MW2O + (16 + lp) * 128 + k]; b1v[1] = sm[MW2O + (16 + lp) * 128 + k + 1];
    d0 = wmma4(a, b0, d0);
    d1 = wmma4(a, b1v, d1);
  }
  float mbias0 = sm[MB2O + lp], mbias1 = sm[MB2O + 16 + lp];
  #pragma unroll
  for (int j = 0; j < 8; ++j) {
    int m = j + 8 * half;
    size_t r = (size_t)r0 + m;
    out_msg[r * 32 + lp]      = d0[j] + mbias0;
    out_msg[r * 32 + 16 + lp] = d1[j] + mbias1;
  }
}

// ---------------- K4: Hebbian trace updates ----------------
__global__ __launch_bounds__(256) void k4_heb(
    const float* __restrict__ msg, const float* __restrict__ prev,
    const float* __restrict__ heb_in, const float* __restrict__ hebb_in,
    const int* __restrict__ conn, const int* __restrict__ bconn,
    float* __restrict__ heb_out, float* __restrict__ hebb_out) {
  int gid = blockIdx.x * blockDim.x + threadIdx.x;
  int wid = gid >> 5, lane = gid & 31;
  int b  = wid >> 14;
  int nl = wid & 16383;
  int nc = nl >> 8, c = nl & 255;
  const float4* m4 = (const float4*)(msg + (size_t)wid * DDim);
  if (lane < 16) {
    int idx = conn[nl * 16 + lane];
    const float4* n4 = (const float4*)(prev + ((size_t)b * NNtot + nc * CCn + idx) * DDim);
    float acc = 0.0f;
    #pragma unroll
    for (int q = 0; q < 8; ++q) {
      float4 mm = m4[q], nn = n4[q];
      acc += mm.x * nn.x + mm.y * nn.y + mm.z * nn.z + mm.w * nn.w;
    }
    size_t o = (size_t)wid * 16 + lane;
    heb_out[o] = 0.9f * heb_in[o] + 0.1f * acc * (1.0f / DDim);
  } else if (c >= AALPHA && c < AALPHA + BBord) {
    int k = lane - 16, cb = c - AALPHA;
    int j = bconn[(nc * BBord + cb) * 16 + k];
    const float4* n4 = (const float4*)(prev + ((size_t)b * NNtot + (j >> 4) * CCn + AALPHA + (j & 15)) * DDim);
    float acc = 0.0f;
    #pragma unroll
    for (int q = 0; q < 8; ++q) {
      float4 mm = m4[q], nn = n4[q];
      acc += mm.x * nn.x + mm.y * nn.y + mm.z * nn.z + mm.w * nn.w;
    }
    size_t o = ((size_t)(b * NCC + nc) * BBord + cb) * 16 + k;
    hebb_out[o] = 0.9f * hebb_in[o] + 0.1f * acc * (1.0f / DDim);
  }
}

// ---------------- K5: readout mean ----------------
__global__ __launch_bounds__(256) void k5_readout(const float* __restrict__ msg,
                                                  float* __restrict__ ro) {
  int t = blockIdx.x * blockDim.x + threadIdx.x;  // 16384
  int d  = t & 31;
  int nc = (t >> 5) & 63;
  int b  = t >> 11;
  float s = 0.0f;
  #pragma unroll
  for (int c = CCn - AALPHA; c < CCn; ++c)
    s += msg[(((size_t)b * NCC + nc) * CCn + c) * DDim + d];
  ro[t] = s * (1.0f / AALPHA);
}

extern "C" void kernel_launch(void* const* d_in, const int* in_sizes, int n_in,
                              void* d_out, int out_size, void* d_ws, size_t ws_size,
                              hipStream_t stream) {
  (void)in_sizes; (void)n_in; (void)out_size; (void)ws_size;
  const float* x     = (const float*)d_in[0];
  const float* h     = (const float*)d_in[1];
  const float* prev  = (const float*)d_in[2];
  const float* dec   = (const float*)d_in[3];
  const float* prim  = (const float*)d_in[4];
  const float* heb   = (const float*)d_in[5];
  const float* hebb  = (const float*)d_in[6];
  const float* sw1   = (const float*)d_in[7];
  const float* sb1   = (const float*)d_in[8];
  const float* sw2   = (const float*)d_in[9];
  const float* sb2   = (const float*)d_in[10];
  const float* mw1   = (const float*)d_in[11];
  const float* mb1   = (const float*)d_in[12];
  const float* mw2   = (const float*)d_in[13];
  const float* mb2   = (const float*)d_in[14];
  const float* modw1 = (const float*)d_in[15];
  const float* modb1 = (const float*)d_in[16];
  const float* modw2 = (const float*)d_in[17];
  const float* modb2 = (const float*)d_in[18];
  const float* nid   = (const float*)d_in[19];
  const int*   conn  = (const int*)d_in[20];
  const int*   bconn = (const int*)d_in[21];

  float* out   = (float*)d_out;
  float* ro    = out;               // (8,2048)
  float* hnew  = out + 16384;       // (8,64,256,32)
  float* msg   = out + 4210688;     // (8,64,256,32)
  float* hebo  = out + 8404992;     // (8,64,256,16)
  float* hebbo = out + 10502144;    // (8,64,16,16)

  float* ws     = (float*)d_ws;
  float* h_upd  = ws;                          // 4194304 floats
  float* modout = ws + 4194304;                // 8519680 floats (8,16384,65)
  float* aggb   = ws + 4194304 + 8519680;      // 4194304 floats

  k0_inject<<<16384, 256, 0, stream>>>(h, x, h_upd);
  k1_mod<<<16384, 32, 0, stream>>>(h_upd, heb, dec, prim, nid,
                                   modw1, modb1, modw2, modb2, modout);
  k2_agg<<<16384, 256, 0, stream>>>(modout, heb, hebb, prev, conn, bconn, aggb);
  k3_mlps<<<1024, 256, 251648, stream>>>(h_upd, aggb, modout, nid,
                                         sw1, sb1, sw2, sb2,
                                         mw1, mb1, mw2, mb2, hnew, msg);
  k4_heb<<<16384, 256, 0, stream>>>(msg, prev, heb, hebb, conn, bconn, hebo, hebbo);
  k5_readout<<<64, 256, 0, stream>>>(msg, ro);
}